// LSTM_3307124818157
// MI455X (gfx1250) — compile-verified
//
#include <hip/hip_runtime.h>
#include <math.h>
#include <stdint.h>

#define T_ 512
#define B_ 2048
#define I_ 128
#define H_ 64
#define V_ 128
#define G_ 256   // 4*H
#define K_ 192   // I+H

typedef __attribute__((ext_vector_type(2))) float v2f;
typedef __attribute__((ext_vector_type(8))) float v8f;
typedef __attribute__((ext_vector_type(4))) unsigned int u32x4;
typedef __attribute__((ext_vector_type(8))) int i32x8;

// ---------------------------------------------------------------------------
// Pack gate weights (concat of Wf,Wif,Wic,Wo -> "Wg"[256,192]) and Wout[128,64]
// into WMMA-B fragment order so the GEMM inner loop does coalesced float2 loads:
//   frag index = ((tile*KC + kc)*32 + lane), value pair = B[k][n], B[k+1][n]
//   with n = tile*16 + (lane&15), k = kc*4 + 2*(lane>>4)   (B[k][n] == W[n][k])
// ---------------------------------------------------------------------------
__global__ void lstm_prep_kernel(const float* __restrict__ Wf,  const float* __restrict__ Wif,
                                 const float* __restrict__ Wic, const float* __restrict__ Wo,
                                 const float* __restrict__ bf,  const float* __restrict__ bif,
                                 const float* __restrict__ bic, const float* __restrict__ bo,
                                 const float* __restrict__ Wout,
                                 float* __restrict__ Bpack, float* __restrict__ bg,
                                 float* __restrict__ OutPack)
{
    const int NG = 16 * 48 * 32;   // gate fragment pairs (16 n-tiles, 48 k-chunks)
    const int NO = 8 * 16 * 32;    // out  fragment pairs (8 n-tiles, 16 k-chunks)
    int idx = blockIdx.x * blockDim.x + threadIdx.x;
    if (idx < NG) {
        int lane = idx & 31;
        int kc   = (idx >> 5) % 48;
        int tile = idx / (48 * 32);
        int n = tile * 16 + (lane & 15);
        int k = kc * 4 + ((lane >> 4) << 1);
        const float* row;
        if (n < 64)       row = Wf  + (size_t)n * K_;
        else if (n < 128) row = Wif + (size_t)(n - 64) * K_;
        else if (n < 192) row = Wic + (size_t)(n - 128) * K_;
        else              row = Wo  + (size_t)(n - 192) * K_;
        Bpack[idx * 2 + 0] = row[k];
        Bpack[idx * 2 + 1] = row[k + 1];
    } else if (idx < NG + NO) {
        int j = idx - NG;
        int lane = j & 31;
        int kc   = (j >> 5) % 16;
        int tile = j / (16 * 32);
        int n = tile * 16 + (lane & 15);
        int k = kc * 4 + ((lane >> 4) << 1);
        OutPack[j * 2 + 0] = Wout[(size_t)n * H_ + k];
        OutPack[j * 2 + 1] = Wout[(size_t)n * H_ + k + 1];
    } else if (idx < NG + NO + G_) {
        int n = idx - NG - NO;
        float v;
        if (n < 64)       v = bf[n];
        else if (n < 128) v = bif[n - 64];
        else if (n < 192) v = bic[n - 128];
        else              v = bo[n - 192];
        bg[n] = v;
    }
}

// ---------------------------------------------------------------------------
// TDM: async DMA of a [2 rows x 128 f32] x-slice into LDS, with LDS padding
// of 4 dwords every 128 dwords -> destination row stride = 132 floats
// (132 % 64 == 4, bank-conflict-free for WMMA A-fragment ds_load_b64s).
// 2D descriptor (D# groups 0,1 only; VADDR2/3 omitted). CDNA5 ISA.
// ---------------------------------------------------------------------------
__device__ __forceinline__ void tdm_load_x_slice(const float* gsrc, unsigned lds_byte_off)
{
    unsigned long long ga = (unsigned long long)(uintptr_t)gsrc;
    u32x4 g0;
    g0[0] = 1u;                                              // count=1, no gather
    g0[1] = lds_byte_off;                                    // lds_addr (bytes)
    g0[2] = (unsigned)(ga & 0xFFFFFFFFull);                  // global_addr[31:0]
    g0[3] = (unsigned)((ga >> 32) & 0x01FFFFFFull)           // global_addr[56:32]
          | (2u << 30);                                      // type=2 ("image")
    i32x8 g1;
    g1[0] = (2 << 16)            // data_size = 4 bytes
          | (1 << 20)            // pad_enable
          | (6 << 22)            // pad_interval: every 128 dwords
          | (3 << 25);           // pad_amount: 4 dwords
    g1[1] = (int)(128u << 16);   // tensor_dim0 = 128 (bits 79:48, low half)
    g1[2] = (int)(2u << 16);     // tensor_dim1 = 2   (bits 111:80, low half)
    g1[3] = (int)(128u << 16);   // tile_dim0 = 128   (bits 127:112)
    g1[4] = 2;                   // tile_dim1 = 2, tile_dim2 = 0
    g1[5] = 128;                 // tensor_dim0_stride = 128 (bits 207:160, low 32)
    g1[6] = 0;                   // stride0 hi / tensor_dim1_stride lo
    g1[7] = 0;                   // tensor_dim1_stride hi
    asm volatile("tensor_load_to_lds %0, %1" :: "s"(g0), "s"(g1) : "memory");
}

// ---------------------------------------------------------------------------
// Persistent recurrence kernel: block owns 16 batch rows for ALL 512 steps
// (LSTM has no cross-batch coupling -> zero inter-block sync needed).
// h/c tiles stay in LDS; x tiles double-buffered via TDM; recurrent-part
// weights pinned in VGPRs; gate GEMM via v_wmma_f32_16x16x4_f32 with the two
// per-wave n-tiles interleaved in one K-loop (shared A-frags, 2 acc chains).
// ---------------------------------------------------------------------------
#define XSTR 132
#define XBUF (16 * XSTR)          // 2112 floats per buffer
#define HOFF (2 * XBUF)           // 4224
#define HSTR 68
#define COFF (HOFF + 16 * HSTR)   // 5312
#define GOFF (COFF + 16 * HSTR)   // 6400
#define GSTR 260
#define SMSZ (GOFF + 16 * GSTR)   // 10560 floats = 42240 bytes

__global__ __launch_bounds__(256) void lstm_seq_kernel(
    const float* __restrict__ x,       // [T,B,I]
    const float* __restrict__ h0,      // [B,H]
    const float* __restrict__ c0,      // [B,H]
    const float* __restrict__ Bpack,   // packed gate weights (fragment order)
    const float* __restrict__ bg,      // [256]
    float* __restrict__ Hs,            // [T,B,H] history of h
    float* __restrict__ hT,            // [B,H] final h -> d_out tail
    float* __restrict__ cT)            // [B,H] final c -> d_out tail
{
    __shared__ float SM[SMSZ];

    const int tid = threadIdx.x;
    const int b0  = blockIdx.x * 16;
    const int wv  = tid >> 5;
    const int lane = tid & 31;
    const int m    = lane & 15;             // A-matrix row for this lane
    const int krow = (lane >> 4) << 1;      // K sub-offset: 0 (lo half) / 2 (hi half)
    const float2* Bp = reinterpret_cast<const float2*>(Bpack);

    const int tile0 = wv * 2;
    const int tile1 = wv * 2 + 1;

    // stage h0/c0 tiles into LDS (16 rows x 64 f32 each)
    {
        int row = tid >> 4;
        int c4  = tid & 15;
        float4 vh = *reinterpret_cast<const float4*>(h0 + (size_t)(b0 + row) * H_ + c4 * 4);
        float4 vc = *reinterpret_cast<const float4*>(c0 + (size_t)(b0 + row) * H_ + c4 * 4);
        float* hp = &SM[HOFF + row * HSTR + c4 * 4];
        float* cp = &SM[COFF + row * HSTR + c4 * 4];
        hp[0] = vh.x; hp[1] = vh.y; hp[2] = vh.z; hp[3] = vh.w;
        cp[0] = vc.x; cp[1] = vc.y; cp[2] = vc.z; cp[3] = vc.w;
    }

    // prologue: each wave DMAs its 2-row x-slice of t=0 into buffer 0
    tdm_load_x_slice(x + ((size_t)0 * B_ + b0 + wv * 2) * I_,
                     (unsigned)((0 * XBUF + wv * 2 * XSTR) * 4));

    // pin the recurrent-part (K=128..191) B-fragments of both n-tiles in VGPRs:
    // reused by all 512 timesteps (2 tiles x 16 chunks x 2 floats = 64 VGPRs)
    float2 bh0[16], bh1[16];
    #pragma unroll
    for (int kc = 0; kc < 16; ++kc) {
        bh0[kc] = Bp[(tile0 * 48 + 32 + kc) * 32 + lane];
        bh1[kc] = Bp[(tile1 * 48 + 32 + kc) * 32 + lane];
    }
    const float bias0 = bg[tile0 * 16 + m];
    const float bias1 = bg[tile1 * 16 + m];
    float* HsBase = Hs + (size_t)b0 * H_;

    for (int t = 0; t < T_; ++t) {
        const int buf = t & 1;
        // prefetch t+1 into the other buffer, then wait for tile t (in-order TDM)
        if (t + 1 < T_) {
            tdm_load_x_slice(x + ((size_t)(t + 1) * B_ + b0 + wv * 2) * I_,
                             (unsigned)(((1 - buf) * XBUF + wv * 2 * XSTR) * 4));
            __builtin_amdgcn_s_wait_tensorcnt(1);
        } else {
            __builtin_amdgcn_s_wait_tensorcnt(0);
        }
        __syncthreads();   // tile t visible to all waves; prev-iter h/c reads done

        // gate GEMM: both n-tiles interleaved in one K-loop (shared A-fragments,
        // two independent accumulator chains hide XDL latency)
        v8f acc0, acc1;
        #pragma unroll
        for (int q = 0; q < 8; ++q) { acc0[q] = bias0; acc1[q] = bias1; }

        const float* Xb = &SM[buf * XBUF + m * XSTR + krow];
        #pragma unroll 4
        for (int kc = 0; kc < 32; ++kc) {          // x part: K = 0..127
            float2 a2 = *reinterpret_cast<const float2*>(Xb + kc * 4);
            float2 b0 = Bp[(tile0 * 48 + kc) * 32 + lane];
            float2 b1 = Bp[(tile1 * 48 + kc) * 32 + lane];
            v2f av; av[0] = a2.x; av[1] = a2.y;
            v2f bv0; bv0[0] = b0.x; bv0[1] = b0.y;
            v2f bv1; bv1[0] = b1.x; bv1[1] = b1.y;
            acc0 = __builtin_amdgcn_wmma_f32_16x16x4_f32(
                false, av, false, bv0, (short)0, acc0, false, false);
            acc1 = __builtin_amdgcn_wmma_f32_16x16x4_f32(
                false, av, false, bv1, (short)0, acc1, false, false);
        }
        const float* Hb = &SM[HOFF + m * HSTR + krow];
        #pragma unroll
        for (int kc = 0; kc < 16; ++kc) {          // h part: K = 128..191 (reg weights)
            float2 a2 = *reinterpret_cast<const float2*>(Hb + kc * 4);
            v2f av; av[0] = a2.x; av[1] = a2.y;
            v2f bv0; bv0[0] = bh0[kc].x; bv0[1] = bh0[kc].y;
            v2f bv1; bv1[0] = bh1[kc].x; bv1[1] = bh1[kc].y;
            acc0 = __builtin_amdgcn_wmma_f32_16x16x4_f32(
                false, av, false, bv0, (short)0, acc0, false, false);
            acc1 = __builtin_amdgcn_wmma_f32_16x16x4_f32(
                false, av, false, bv1, (short)0, acc1, false, false);
        }
        {
            const int rbase = (lane >> 4) << 3;    // C layout: vgpr q -> rows q / q+8
            #pragma unroll
            for (int q = 0; q < 8; ++q) {
                SM[GOFF + (rbase + q) * GSTR + tile0 * 16 + m] = acc0[q];
                SM[GOFF + (rbase + q) * GSTR + tile1 * 16 + m] = acc1[q];
            }
        }
        __syncthreads();   // gate pre-activations ready

        // elementwise cell update: 16 rows x 64 hidden = 1024 elems, 4/thread
        float* HsT = HsBase + (size_t)t * B_ * H_;
        #pragma unroll
        for (int i = 0; i < 4; ++i) {
            int idx = tid + i * 256;
            int row = idx >> 6;
            int j   = idx & 63;
            const float* g = &SM[GOFF + row * GSTR];
            float f  = 1.0f / (1.0f + expf(-g[j]));
            float ig = 1.0f / (1.0f + expf(-g[64 + j]));
            float cd = tanhf(g[128 + j]);
            float o  = 1.0f / (1.0f + expf(-g[192 + j]));
            float cn = SM[COFF + row * HSTR + j] * f + cd * ig;
            float hn = tanhf(cn) * o;
            SM[COFF + row * HSTR + j] = cn;
            SM[HOFF + row * HSTR + j] = hn;
            HsT[row * H_ + j] = hn;
        }
        __syncthreads();   // h/c updated before next step's GEMM reads them
    }

    // write final h/c straight to d_out tail
    {
        int row = tid >> 4;
        int c4  = tid & 15;
        float* hp = hT + (size_t)(b0 + row) * H_ + c4 * 4;
        float* cp = cT + (size_t)(b0 + row) * H_ + c4 * 4;
        #pragma unroll
        for (int k = 0; k < 4; ++k) {
            hp[k] = SM[HOFF + row * HSTR + c4 * 4 + k];
            cp[k] = SM[COFF + row * HSTR + c4 * 4 + k];
        }
    }
}

// ---------------------------------------------------------------------------
// Output projection + softmax over all T*B rows (fully parallel).
// Block owns 16 rows; 8 waves each do one 16x16 n-tile (V=128), K=64.
// ---------------------------------------------------------------------------
__global__ __launch_bounds__(256) void lstm_out_kernel(
    const float* __restrict__ Hs,       // [T*B, H]
    const float* __restrict__ OutPack,  // packed Wout fragments
    const float* __restrict__ bout,     // [V]
    float* __restrict__ probs)          // [T*B, V]
{
    __shared__ float Ht[16][68];        // 68%64==4 -> conflict-free
    __shared__ float Ls[16][132];       // logits, 132%64==4

    const int tid = threadIdx.x;
    const size_t r0 = (size_t)blockIdx.x * 16;

    {
        int row = tid >> 4;
        int c4  = tid & 15;
        float4 v = *reinterpret_cast<const float4*>(Hs + (r0 + row) * H_ + c4 * 4);
        Ht[row][c4 * 4 + 0] = v.x; Ht[row][c4 * 4 + 1] = v.y;
        Ht[row][c4 * 4 + 2] = v.z; Ht[row][c4 * 4 + 3] = v.w;
    }
    __syncthreads();

    const int wave = tid >> 5;
    const int lane = tid & 31;
    const int m    = lane & 15;
    const int krow = (lane >> 4) << 1;
    const int n0   = wave * 16;
    const float2* Op = reinterpret_cast<const float2*>(OutPack);

    {
        const float bias = bout[n0 + m];
        v8f acc;
        #pragma unroll
        for (int q = 0; q < 8; ++q) acc[q] = bias;
        #pragma unroll
        for (int kc = 0; kc < 16; ++kc) {  // K = 64
            float2 a2 = *reinterpret_cast<const float2*>(&Ht[m][kc * 4 + krow]);
            float2 b2 = Op[(wave * 16 + kc) * 32 + lane];
            v2f av; av[0] = a2.x; av[1] = a2.y;
            v2f bv; bv[0] = b2.x; bv[1] = b2.y;
            acc = __builtin_amdgcn_wmma_f32_16x16x4_f32(
                false, av, false, bv, (short)0, acc, false, false);
        }
        const int rbase = (lane >> 4) << 3;
        #pragma unroll
        for (int q = 0; q < 8; ++q)
            Ls[rbase + q][n0 + m] = acc[q];
    }
    __syncthreads();

    // softmax over 128 cols, 16 lanes per row, 8 cols per lane
    const int row = tid >> 4;
    const int l16 = tid & 15;
    float v[8];
    float mx = -INFINITY;
    #pragma unroll
    for (int i = 0; i < 8; ++i) { v[i] = Ls[row][l16 + 16 * i]; mx = fmaxf(mx, v[i]); }
    #pragma unroll
    for (int off = 8; off >= 1; off >>= 1) mx = fmaxf(mx, __shfl_xor(mx, off, 16));
    float s = 0.0f;
    #pragma unroll
    for (int i = 0; i < 8; ++i) { v[i] = expf(v[i] - mx); s += v[i]; }
    #pragma unroll
    for (int off = 8; off >= 1; off >>= 1) s += __shfl_xor(s, off, 16);
    const float inv = 1.0f / s;
    float* pout = probs + (r0 + row) * (size_t)V_;
    #pragma unroll
    for (int i = 0; i < 8; ++i) pout[l16 + 16 * i] = v[i] * inv;
}

// ---------------------------------------------------------------------------
extern "C" void kernel_launch(void* const* d_in, const int* in_sizes, int n_in,
                              void* d_out, int out_size, void* d_ws, size_t ws_size,
                              hipStream_t stream)
{
    const float* x    = (const float*)d_in[0];
    const float* h0   = (const float*)d_in[1];
    const float* c0   = (const float*)d_in[2];
    const float* Wf   = (const float*)d_in[3];
    const float* bf   = (const float*)d_in[4];
    const float* Wif  = (const float*)d_in[5];
    const float* bif  = (const float*)d_in[6];
    const float* Wic  = (const float*)d_in[7];
    const float* bic  = (const float*)d_in[8];
    const float* Wo   = (const float*)d_in[9];
    const float* bo   = (const float*)d_in[10];
    const float* Wout = (const float*)d_in[11];
    const float* bout = (const float*)d_in[12];
    float* out = (float*)d_out;

    char* ws = (char*)d_ws;
    size_t off = 0;
    float* Bpack   = (float*)(ws + off); off += (size_t)G_ * K_ * 4;
    float* bg      = (float*)(ws + off); off += 1024;
    float* OutPack = (float*)(ws + off); off += (size_t)V_ * H_ * 4;
    float* Hs      = (float*)(ws + off); off += (size_t)T_ * B_ * H_ * 4;

    const int prepN = 16 * 48 * 32 + 8 * 16 * 32 + G_;
    lstm_prep_kernel<<<(prepN + 255) / 256, 256, 0, stream>>>(
        Wf, Wif, Wic, Wo, bf, bif, bic, bo, Wout, Bpack, bg, OutPack);

    float* hT = out + (size_t)T_ * B_ * V_;
    float* cT = hT + (size_t)B_ * H_;

    lstm_seq_kernel<<<B_ / 16, 256, 0, stream>>>(x, h0, c0, Bpack, bg, Hs, hT, cT);

    lstm_out_kernel<<<(T_ * B_) / 16, 256, 0, stream>>>(Hs, OutPack, bout, out);
}